// EmbedLinear_75720273428632
// MI455X (gfx1250) — compile-verified
//
#include <hip/hip_runtime.h>

// Problem constants (match reference).
#define NNZ    65536
#define NCH    16384       // children (rows of sparse W)
#define WSZ    2048        // weight size = x columns = parents
#define BATCH  2048
#define OUTW   (WSZ + NCH) // 18432 output columns
#define RPT    8           // batch rows staged per tile (8 * 2048 * 4B = 64 KB LDS)
#define CTILE  256         // children per inner tile (== blockDim.x)
#define NCT    (NCH / CTILE)  // 64 child tiles total
#define CGROUPS 4          // blocks in x-dim; each block loops NCT/CGROUPS = 16 ctiles
                           // staging traffic = CGROUPS * 16.8 MB; edge traffic fixed.

typedef __attribute__((ext_vector_type(4))) unsigned int v4u;
typedef __attribute__((ext_vector_type(8))) int v8i;
typedef __attribute__((ext_vector_type(4))) int v4i;

// ---------------- CSR construction (device-side, per call) ----------------

__global__ void k_zero(int* __restrict__ p, int n) {
    int i = blockIdx.x * blockDim.x + threadIdx.x;
    if (i < n) p[i] = 0;
}

__global__ void k_hist(const int* __restrict__ child, int* __restrict__ counts) {
    int e = blockIdx.x * blockDim.x + threadIdx.x;
    if (e < NNZ) atomicAdd(&counts[child[e]], 1);
}

// Single-block exclusive scan over NCH counters (16 chunks of 1024, Hillis-Steele).
__global__ void k_scan(const int* __restrict__ counts,
                       int* __restrict__ offs,
                       int* __restrict__ cursors) {
    __shared__ int sh[1024];
    __shared__ int carry;
    if (threadIdx.x == 0) carry = 0;
    __syncthreads();
    for (int base = 0; base < NCH; base += 1024) {
        int v = counts[base + threadIdx.x];
        sh[threadIdx.x] = v;
        __syncthreads();
        for (int off = 1; off < 1024; off <<= 1) {
            int t = (threadIdx.x >= off) ? sh[threadIdx.x - off] : 0;
            __syncthreads();
            sh[threadIdx.x] += t;
            __syncthreads();
        }
        int incl = sh[threadIdx.x];
        int c = carry;
        offs[base + threadIdx.x]    = c + incl - v;
        cursors[base + threadIdx.x] = c + incl - v;
        __syncthreads();
        if (threadIdx.x == 1023) carry = c + incl;
        __syncthreads();
    }
    if (threadIdx.x == 0) offs[NCH] = carry;   // == NNZ
}

// Bucket edges by child; pack (parent, value) into one int2 for b64 loads later.
__global__ void k_scatter(const int* __restrict__ child,
                          const int* __restrict__ parent,
                          const float* __restrict__ vals,
                          int* __restrict__ cursors,
                          int2* __restrict__ se) {
    int e = blockIdx.x * blockDim.x + threadIdx.x;
    if (e < NNZ) {
        int c = child[e];
        int pos = atomicAdd(&cursors[c], 1);
        se[pos] = make_int2(parent[e], __float_as_int(vals[e]));
    }
}

// ---------------- out[:, :WSZ] = x (vectorized copy) ----------------

__global__ void k_copy(const float4* __restrict__ x4, float* __restrict__ out) {
    int i = blockIdx.x * blockDim.x + threadIdx.x;     // over BATCH*WSZ/4
    int row = i / (WSZ / 4);
    int col = i % (WSZ / 4);
    ((float4*)(out + (size_t)row * OUTW))[col] = x4[i];
}

// ---------------- main: CSR gather from TDM-staged x tile ----------------
// grid = (CGROUPS, BATCH/RPT), block = CTILE threads, dynLDS = RPT*WSZ*4 B.
// One TDM stage per block, amortized over NCT/CGROUPS child tiles.

__global__ void k_spmm(const float* __restrict__ x,
                       const int* __restrict__ offs,
                       const int2* __restrict__ se,
                       float* __restrict__ out) {
    extern __shared__ float smem[];                    // [RPT][WSZ]
    const int b0 = blockIdx.y * RPT;

#if defined(__gfx1250__) && __has_builtin(__builtin_amdgcn_tensor_load_to_lds)
    // Tensor Data Mover: DMA x[b0:b0+RPT, 0:WSZ] (2D tile) into LDS.
    // One wave issues the descriptor; TENSORcnt tracked by issuing wave.
    if (threadIdx.x < 32) {
        unsigned lds_base = (unsigned)(unsigned long long)(uintptr_t)smem;
        unsigned long long ga =
            (unsigned long long)(uintptr_t)(x + (size_t)b0 * WSZ);
        v4u g0;
        g0.x = 1u;                                        // count=1 (valid user D#)
        g0.y = lds_base;                                  // lds_addr (bytes)
        g0.z = (unsigned)ga;                              // global_addr[31:0]
        g0.w = ((unsigned)(ga >> 32) & 0x01FFFFFFu)       // global_addr[56:32]
               | (2u << 30);                              // type = 2 ("image")
        v8i g1;
        g1[0] = 0x00020000;                               // data_size=2 (4 bytes)
        g1[1] = (int)((WSZ & 0xFFFF) << 16);              // tensor_dim0 lo16
        g1[2] = (int)(((WSZ >> 16) & 0xFFFF)              // tensor_dim0 hi16
               | ((BATCH & 0xFFFF) << 16));               // tensor_dim1 lo16
        g1[3] = (int)(((BATCH >> 16) & 0xFFFF)            // tensor_dim1 hi16
               | ((WSZ & 0xFFFF) << 16));                 // tile_dim0 = WSZ
        g1[4] = RPT;                                      // tile_dim1=RPT, tile_dim2=0
        g1[5] = WSZ;                                      // tensor_dim0_stride lo32
        g1[6] = 0;                                        // stride hi16 | dim1_stride lo16
        g1[7] = 0;
        v4i g2 = {0, 0, 0, 0};
        v4i g3 = {0, 0, 0, 0};
#if defined(__clang_major__) && __clang_major__ >= 23
        v8i g4 = {0, 0, 0, 0, 0, 0, 0, 0};
        __builtin_amdgcn_tensor_load_to_lds(g0, g1, g2, g3, g4, 0);
#else
        __builtin_amdgcn_tensor_load_to_lds(g0, g1, g2, g3, 0);
#endif
#if __has_builtin(__builtin_amdgcn_s_wait_tensorcnt)
        __builtin_amdgcn_s_wait_tensorcnt(0);
#else
        asm volatile("s_wait_tensorcnt 0" ::: "memory");
#endif
    }
#else
    // Fallback: cooperative contiguous copy (rows b0..b0+RPT-1 are contiguous).
    {
        const float4* src = (const float4*)(x + (size_t)b0 * WSZ);
        float4* dst = (float4*)smem;
        for (int i = threadIdx.x; i < RPT * WSZ / 4; i += blockDim.x)
            dst[i] = src[i];
    }
#endif
    __syncthreads();

    // Amortize the staged tile over several child tiles (no further barriers:
    // the loop body only reads LDS).
    for (int ct = blockIdx.x; ct < NCT; ct += CGROUPS) {
        const int c   = ct * CTILE + (int)threadIdx.x;
        const int beg = offs[c];
        const int end = offs[c + 1];

#if defined(__gfx1250__)
        __builtin_prefetch(&se[beg], 0, 1);               // global_prefetch_b8
#endif

        float acc[RPT];
#pragma unroll
        for (int r = 0; r < RPT; ++r) acc[r] = 0.0f;

        for (int j = beg; j < end; ++j) {
            const int2  e = se[j];                        // one b64 load per edge
            const int   p = e.x;
            const float v = __int_as_float(e.y);
            const float* xl = smem + p;                   // LDS gather
#pragma unroll
            for (int r = 0; r < RPT; ++r)
                acc[r] += v * xl[r * WSZ];
        }

        // Write-once, coalesced across threads (contiguous in c).
#pragma unroll
        for (int r = 0; r < RPT; ++r)
            out[(size_t)(b0 + r) * OUTW + WSZ + c] = acc[r];
    }
}

// ---------------- launch ----------------

extern "C" void kernel_launch(void* const* d_in, const int* in_sizes, int n_in,
                              void* d_out, int out_size, void* d_ws, size_t ws_size,
                              hipStream_t stream) {
    const float* x      = (const float*)d_in[0];
    const float* vals   = (const float*)d_in[1];
    const int*   child  = (const int*)d_in[2];
    const int*   parent = (const int*)d_in[3];
    float*       out    = (float*)d_out;

    // Workspace layout (ints): counts[NCH] | offs[NCH+1] | cursors[NCH] | pad | se[NNZ]x2
    int*   counts  = (int*)d_ws;
    int*   offs    = counts + NCH;
    int*   cursors = offs + NCH + 1;
    int2*  se      = (int2*)(cursors + NCH + 1);   // +1 keeps int2 8B-aligned

    k_zero   <<<NCH / 256, 256, 0, stream>>>(counts, NCH);
    k_hist   <<<NNZ / 256, 256, 0, stream>>>(child, counts);
    k_scan   <<<1, 1024, 0, stream>>>(counts, offs, cursors);
    k_scatter<<<NNZ / 256, 256, 0, stream>>>(child, parent, vals, cursors, se);
    k_copy   <<<(BATCH * WSZ / 4) / 256, 256, 0, stream>>>((const float4*)x, out);

    dim3 grid(CGROUPS, BATCH / RPT);
    k_spmm<<<grid, CTILE, RPT * WSZ * sizeof(float), stream>>>(x, offs, se, out);
}